// AttentionPooling_34703335752445
// MI455X (gfx1250) — compile-verified
//
#include <hip/hip_runtime.h>
#include <math.h>

typedef __attribute__((ext_vector_type(2))) float v2f;
typedef __attribute__((ext_vector_type(8))) float v8f;

#define NSEG 1024

// ---- monotone float <-> uint key (memset-0 == -infinity identity for unsigned max)
__device__ __forceinline__ unsigned fkey(float f) {
    unsigned u = __float_as_uint(f);
    return (u & 0x80000000u) ? ~u : (u | 0x80000000u);
}
__device__ __forceinline__ float fdec(unsigned k) {
    unsigned u = (k & 0x80000000u) ? (k ^ 0x80000000u) : ~k;
    return __uint_as_float(u);
}

// ============================================================================
// Kernel 1: h = gelu(x @ W1 + b1); scores = h @ W2 + b2
// 256 threads = 8 waves; each wave computes a 16-row strip with
// V_WMMA_F32_16X16X4_F32. W1 is staged once per block into LDS, pre-swizzled
// into the exact B-fragment layout, so the inner loop is ds_load_b64 + wmma.
// ============================================================================
__global__ __launch_bounds__(256)
void mlp_scores_kernel(const float* __restrict__ x,
                       const float* __restrict__ W1,
                       const float* __restrict__ b1,
                       const float* __restrict__ W2,
                       const float* __restrict__ b2,
                       float* __restrict__ scores,
                       int N)
{
    // B-fragment pack: w1pack[(nt*32 + kt)*32 + lane] =
    //   { W1[4kt + 2*(lane/16)    ][nt*16 + lane%16],
    //     W1[4kt + 2*(lane/16) + 1][nt*16 + lane%16] }
    __shared__ v2f w1pack[8 * 32 * 32];   // 64 KB

    const int tid  = threadIdx.x;
    const int lane = tid & 31;
    const int wave = tid >> 5;
    const int lmod = lane & 15;
    const int lhi  = lane >> 4;           // 0 or 1
    const int rowbase = blockIdx.x * 128 + wave * 16;

    // ---- cooperative W1 pack: coalesced global reads, swizzled LDS writes
    {
        float* dst = (float*)w1pack;
        for (int idx = tid; idx < 128 * 128; idx += 256) {
            const int k = idx >> 7;       // row of W1 (K dim)
            const int n = idx & 127;      // col of W1 (N dim)
            const int nt = n >> 4, lm = n & 15;
            const int kt = k >> 2, lh = (k >> 1) & 1, r = k & 1;
            dst[(((nt * 32 + kt) * 32 + lh * 16 + lm) << 1) + r] = W1[idx];
        }
    }

    // ---- preload A fragments (this lane's x row, 64 floats as 32 float2)
    int arow = rowbase + lmod;
    if (arow >= N) arow = N - 1;          // clamp; stores are guarded below
    const float* __restrict__ xrow = x + (size_t)arow * 128;

    v2f afrag[32];
#pragma unroll
    for (int kt = 0; kt < 32; ++kt)
        afrag[kt] = *(const v2f*)(xrow + kt * 4 + lhi * 2);

    __syncthreads();

    float part[8];
#pragma unroll
    for (int r = 0; r < 8; ++r) part[r] = 0.0f;

    const v2f* __restrict__ bbase = w1pack + lane;

    for (int nt = 0; nt < 8; ++nt) {
        const int n = nt * 16 + lmod;     // B/C column this lane owns
        v8f acc = {};
#pragma unroll
        for (int kt = 0; kt < 32; ++kt) {
            v2f b = bbase[(nt * 32 + kt) * 32];   // ds_load_b64, imm offset
            acc = __builtin_amdgcn_wmma_f32_16x16x4_f32(
                false, afrag[kt], false, b, (short)0, acc, false, false);
        }
        const float bias = b1[n];
        const float w2n  = W2[n];
        // C/D layout: VGPR r, lane l -> [M = r + 8*(l/16)][N = l%16]
#pragma unroll
        for (int r = 0; r < 8; ++r) {
            float h = acc[r] + bias;
            float g = 0.5f * h * (1.0f + erff(h * 0.70710678118654752f)); // exact GELU
            part[r] = fmaf(g, w2n, part[r]);
        }
    }

    const float beta = b2[0];
#pragma unroll
    for (int r = 0; r < 8; ++r) {
        float v = part[r];
        v += __shfl_xor(v, 1, 16);
        v += __shfl_xor(v, 2, 16);
        v += __shfl_xor(v, 4, 16);
        v += __shfl_xor(v, 8, 16);
        if (lmod == 0) {
            int row = rowbase + r + 8 * lhi;
            if (row < N) scores[row] = v + beta;
        }
    }
}

// ============================================================================
// Kernel 2: per-segment max (batch is sorted -> wave-aggregate atomics)
// ============================================================================
__global__ __launch_bounds__(256)
void segmax_kernel(const float* __restrict__ scores,
                   const int* __restrict__ batch,
                   unsigned* __restrict__ segkey, int N)
{
    int i = blockIdx.x * blockDim.x + threadIdx.x;
    if (i >= N) return;
    int b = batch[i];
    unsigned key = fkey(scores[i]);
    bool full = ((i & ~31) + 31) < N;            // whole wave active?
    if (full) {
        int b0 = __shfl(b, 0);
        if (__all(b == b0)) {
            unsigned k = key;
            k = max(k, (unsigned)__shfl_xor((int)k, 1));
            k = max(k, (unsigned)__shfl_xor((int)k, 2));
            k = max(k, (unsigned)__shfl_xor((int)k, 4));
            k = max(k, (unsigned)__shfl_xor((int)k, 8));
            k = max(k, (unsigned)__shfl_xor((int)k, 16));
            if ((threadIdx.x & 31) == 0) atomicMax(&segkey[b0], k);
            return;
        }
    }
    atomicMax(&segkey[b], key);
}

// ============================================================================
// Kernel 3: e = exp(s - segmax); denom += e   (e overwrites scores buffer)
// ============================================================================
__global__ __launch_bounds__(256)
void segexp_kernel(float* __restrict__ scores,
                   const int* __restrict__ batch,
                   const unsigned* __restrict__ segkey,
                   float* __restrict__ denom, int N)
{
    int i = blockIdx.x * blockDim.x + threadIdx.x;
    if (i >= N) return;
    int b = batch[i];
    float e = __expf(scores[i] - fdec(segkey[b]));
    scores[i] = e;
    bool full = ((i & ~31) + 31) < N;
    if (full) {
        int b0 = __shfl(b, 0);
        if (__all(b == b0)) {
            float s = e;
            s += __shfl_xor(s, 1);
            s += __shfl_xor(s, 2);
            s += __shfl_xor(s, 4);
            s += __shfl_xor(s, 8);
            s += __shfl_xor(s, 16);
            if ((threadIdx.x & 31) == 0) atomicAdd(&denom[b0], s);
            return;
        }
    }
    atomicAdd(&denom[b], e);
}

// ============================================================================
// Kernel 4: inv_denom = 1/denom (so the 2M x 128 pooling pass multiplies)
// ============================================================================
__global__ void seginv_kernel(const float* __restrict__ denom,
                              float* __restrict__ invd)
{
    int i = blockIdx.x * blockDim.x + threadIdx.x;
    if (i < NSEG) invd[i] = 1.0f / denom[i];
}

// ============================================================================
// Kernel 5: pooled[seg,:] += (e/denom) * x[i,:]; one thread per column,
// sorted rows -> local accumulate, atomic flush only on segment boundary.
// ============================================================================
#define POOL_ROWS 512
__global__ __launch_bounds__(128)
void pool_kernel(const float* __restrict__ x,
                 const int* __restrict__ batch,
                 const float* __restrict__ e,
                 const float* __restrict__ invd,
                 float* __restrict__ pooled, int N)
{
    const int d = threadIdx.x;                       // 0..127
    int r0 = blockIdx.x * POOL_ROWS;
    if (r0 >= N) return;
    int r1 = r0 + POOL_ROWS; if (r1 > N) r1 = N;

    int curseg = batch[r0];
    float acc = 0.0f;
    for (int i = r0; i < r1; ++i) {
        int b = batch[i];                            // broadcast load
        if (b != curseg) {
            atomicAdd(&pooled[(size_t)curseg * 128 + d], acc);
            acc = 0.0f;
            curseg = b;
        }
        float w = e[i] * invd[b];                    // broadcast loads
        acc = fmaf(w, x[(size_t)i * 128 + d], acc);  // coalesced 512B row
    }
    atomicAdd(&pooled[(size_t)curseg * 128 + d], acc);
}

// ============================================================================
extern "C" void kernel_launch(void* const* d_in, const int* in_sizes, int n_in,
                              void* d_out, int out_size, void* d_ws, size_t ws_size,
                              hipStream_t stream)
{
    const float* x     = (const float*)d_in[0];
    const int*   batch = (const int*)  d_in[1];
    const float* W1    = (const float*)d_in[2];
    const float* b1    = (const float*)d_in[3];
    const float* W2    = (const float*)d_in[4];
    const float* b2    = (const float*)d_in[5];
    float* pooled = (float*)d_out;
    const int N = in_sizes[1];

    // workspace layout: [scores/e : N floats][segkey : 1024 u32][denom : 1024 f32][invd : 1024 f32]
    float*    scores = (float*)d_ws;
    size_t    off    = (((size_t)N * sizeof(float)) + 255) & ~(size_t)255;
    unsigned* segkey = (unsigned*)((char*)d_ws + off);
    float*    denom  = (float*)(segkey + NSEG);
    float*    invd   = denom + NSEG;

    // segkey/denom/invd are contiguous: zero all three (key 0 == -inf identity)
    hipMemsetAsync(segkey, 0, 3 * NSEG * sizeof(unsigned), stream);
    hipMemsetAsync(pooled, 0, (size_t)out_size * sizeof(float), stream);

    mlp_scores_kernel<<<(N + 127) / 128, 256, 0, stream>>>(x, W1, b1, W2, b2, scores, N);
    segmax_kernel<<<(N + 255) / 256, 256, 0, stream>>>(scores, batch, segkey, N);
    segexp_kernel<<<(N + 255) / 256, 256, 0, stream>>>(scores, batch, segkey, denom, N);
    seginv_kernel<<<(NSEG + 255) / 256, 256, 0, stream>>>(denom, invd);
    pool_kernel<<<(N + POOL_ROWS - 1) / POOL_ROWS, 128, 0, stream>>>(x, batch, scores, invd, pooled, N);
}